// TransMatch_74689481278072
// MI455X (gfx1250) — compile-verified
//
#include <hip/hip_runtime.h>

// ---------------------------------------------------------------------------
// Problem constants (from the reference)
// ---------------------------------------------------------------------------
#define BS   256
#define P    64
#define NS   32
#define EMB  128

#define M1   (BS * P * 2)   // 32768 rows of GEMM1 / neighbor_vectors
#define M2   (BS * P)       // 16384 rows of GEMM2 / sv
#define NCOL 128            // output columns of both GEMMs

typedef __attribute__((ext_vector_type(2))) float v2f;
typedef __attribute__((ext_vector_type(4))) float v4f;
typedef __attribute__((ext_vector_type(8))) float v8f;

// ---------------------------------------------------------------------------
// Kernel 1: masked mean over NS samples.  Bandwidth-dominant pass (537 MB in).
// One wave per (b,p,j) row.  Each lane owns 4 consecutive emb elements ->
// global_load_b128, 512B fully-coalesced per wave per sample.
// ---------------------------------------------------------------------------
__global__ void __launch_bounds__(256)
agg_kernel(const float* __restrict__ edge,   // [M1][NS][EMB]
           const float* __restrict__ masks,  // [M1][NS]
           float*       __restrict__ agg)    // [M1][EMB]
{
    const int wave = (int)((blockIdx.x * blockDim.x + threadIdx.x) >> 5);
    const int lane = (int)(threadIdx.x & 31);

    const float* __restrict__ erow = edge + (size_t)wave * (NS * EMB) + lane * 4;
    const float* __restrict__ mrow = masks + (size_t)wave * NS;

    v4f   acc = {0.f, 0.f, 0.f, 0.f};
    float cnt = 0.f;
#pragma unroll 4
    for (int s = 0; s < NS; ++s) {
        const float m = mrow[s];
        const v4f   e = *(const v4f*)(erow + s * EMB);
        acc += e * m;
        cnt += m;
    }
    const float nn = (cnt == 0.f) ? 1.f : cnt;
    v4f r;
    r.x = acc.x / nn; r.y = acc.y / nn; r.z = acc.z / nn; r.w = acc.w / nn;
    *(v4f*)(agg + (size_t)wave * EMB + lane * 4) = r;
}

// ---------------------------------------------------------------------------
// WMMA f32 16x16x4 K-loop over a klen-element K slice.
//   A fragment (16x4, MxK): lane l (m=l&15, h=l>>4): {A[m][k+2h], A[m][k+2h+1]}
//   B fragment (4x16, KxN): lane l (n=l&15, h=l>>4): {W[k+2h][n], W[k+2h+1][n]}
// W has row stride NCOL (=128).  kk is always even -> float2 loads are aligned.
// ---------------------------------------------------------------------------
__device__ __forceinline__ v8f
tile_accum(const float* __restrict__ A, int lda, int klen,
           const float* __restrict__ W,
           int r, int h, int ncol, v8f acc)
{
#pragma unroll 4
    for (int k = 0; k < klen; k += 4) {
        const int kk = k + (h << 1);
        const v2f a = *(const v2f*)(A + (size_t)r * lda + kk);
        v2f b;
        b.x = W[(size_t)kk * NCOL + ncol];
        b.y = W[(size_t)kk * NCOL + NCOL + ncol];
        acc = __builtin_amdgcn_wmma_f32_16x16x4_f32(
            /*neg_a=*/false, a, /*neg_b=*/false, b,
            /*c_mod=*/(short)0, acc, /*reuse_a=*/false, /*reuse_b=*/false);
    }
    return acc;
}

__device__ __forceinline__ void
tile_store(float* __restrict__ out, const float* __restrict__ bias,
           int m0, int ntile, int lane, v8f acc)
{
    const int h  = lane >> 4;
    const int n  = ntile * 16 + (lane & 15);
    const float bv = bias[n];
#pragma unroll
    for (int j = 0; j < 8; ++j) {
        const int m = m0 + j + 8 * h;
        out[(size_t)m * NCOL + n] = acc[j] + bv;
    }
}

// ---------------------------------------------------------------------------
// Kernel 2: neighbor_vectors = [ent | edge_agg] @ W_ent + b_ent
//   M=32768, K=256, N=128.  One wave per 16x16 tile; 8 ntiles x 2048 mtiles.
// ---------------------------------------------------------------------------
__global__ void __launch_bounds__(256)
gemm_nv_kernel(const float* __restrict__ ent,   // [M1][EMB]
               const float* __restrict__ agg,   // [M1][EMB]
               const float* __restrict__ W,     // [2*EMB][NCOL]
               const float* __restrict__ bias,  // [NCOL]
               float*       __restrict__ out)   // [M1][NCOL]
{
    const int wid   = (int)((blockIdx.x * blockDim.x + threadIdx.x) >> 5);
    const int lane  = (int)(threadIdx.x & 31);
    const int ntile = wid & 7;
    const int mtile = wid >> 3;
    const int m0    = mtile * 16;
    const int r     = lane & 15;
    const int h     = lane >> 4;
    const int ncol  = ntile * 16 + r;

    v8f acc = {};
    // K = 0..127  : neighbor_entity rows (stride EMB), W rows 0..127
    acc = tile_accum(ent + (size_t)m0 * EMB, EMB, 128, W, r, h, ncol, acc);
    // K = 128..255: edge_agg rows (stride EMB), W rows 128..255
    acc = tile_accum(agg + (size_t)m0 * EMB, EMB, 128, W + 128 * NCOL, r, h, ncol, acc);

    tile_store(out, bias, m0, ntile, lane, acc);
}

// ---------------------------------------------------------------------------
// Kernel 3: sv = [self | nv_view] @ W_out + b_out
//   M=16384, K=384, N=128.  nv_view is neighbor_vectors reinterpreted as
//   [M2][2*EMB] (same flat layout), so it is read with row stride 256.
// ---------------------------------------------------------------------------
__global__ void __launch_bounds__(256)
gemm_sv_kernel(const float* __restrict__ selfv, // [M2][EMB]
               const float* __restrict__ nv,    // [M2][2*EMB]
               const float* __restrict__ W,     // [3*EMB][NCOL]
               const float* __restrict__ bias,  // [NCOL]
               float*       __restrict__ out)   // [M2][NCOL]
{
    const int wid   = (int)((blockIdx.x * blockDim.x + threadIdx.x) >> 5);
    const int lane  = (int)(threadIdx.x & 31);
    const int ntile = wid & 7;
    const int mtile = wid >> 3;
    const int m0    = mtile * 16;
    const int r     = lane & 15;
    const int h     = lane >> 4;
    const int ncol  = ntile * 16 + r;

    v8f acc = {};
    // K = 0..127  : self rows (stride EMB), W rows 0..127
    acc = tile_accum(selfv + (size_t)m0 * EMB, EMB, 128, W, r, h, ncol, acc);
    // K = 128..383: nv rows (stride 2*EMB), W rows 128..383
    acc = tile_accum(nv + (size_t)m0 * (2 * EMB), 2 * EMB, 256, W + 128 * NCOL,
                     r, h, ncol, acc);

    tile_store(out, bias, m0, ntile, lane, acc);
}

// ---------------------------------------------------------------------------
// Launch: three kernels on `stream`.
//   d_out = [ sv (M2*128 floats) | neighbor_vectors (M1*128 floats) ]
//   d_ws  = edge_agg scratch (M1*128 floats = 16.8 MB)
// ---------------------------------------------------------------------------
extern "C" void kernel_launch(void* const* d_in, const int* in_sizes, int n_in,
                              void* d_out, int out_size, void* d_ws, size_t ws_size,
                              hipStream_t stream) {
    const float* selfv = (const float*)d_in[0];  // [BS,P,EMB]
    const float* ent   = (const float*)d_in[1];  // [BS,P,2,EMB]
    const float* edge  = (const float*)d_in[2];  // [BS,P,2,NS,EMB]
    const float* masks = (const float*)d_in[3];  // [BS,P,2,NS,1]
    const float* W_ent = (const float*)d_in[4];  // [2*EMB,EMB]
    const float* b_ent = (const float*)d_in[5];  // [EMB]
    const float* W_out = (const float*)d_in[6];  // [3*EMB,EMB]
    const float* b_out = (const float*)d_in[7];  // [EMB]

    float* sv  = (float*)d_out;                       // output 0
    float* nv  = (float*)d_out + (size_t)M2 * NCOL;   // output 1
    float* agg = (float*)d_ws;                        // scratch edge_agg

    // 1) masked mean: M1 waves, 8 waves/block
    agg_kernel<<<M1 / 8, 256, 0, stream>>>(edge, masks, agg);

    // 2) GEMM1: (M1/16)*8 tiles, 1 wave each, 8 waves/block
    gemm_nv_kernel<<<(M1 / 16) * 8 / 8, 256, 0, stream>>>(ent, agg, W_ent, b_ent, nv);

    // 3) GEMM2: (M2/16)*8 tiles
    gemm_sv_kernel<<<(M2 / 16) * 8 / 8, 256, 0, stream>>>(selfv, nv, W_out, b_out, sv);
}